// Decoders_4028679324290
// MI455X (gfx1250) — compile-verified
//
#include <hip/hip_runtime.h>
#include <math.h>

#define RDIM 256
#define CDIM 32
#define NSEG 8

typedef __attribute__((ext_vector_type(2))) float v2f;
typedef __attribute__((ext_vector_type(8))) float v8f;

__device__ __forceinline__ float sigmoidf_(float x) { return 1.0f / (1.0f + expf(-x)); }

// Accumulate one plane's bilerp into the wave's A-layout channel pairs.
// Lane half h owns channels {4k+2h, 4k+2h+1} for k = 0..7 (matches the
// 16x4 f32 A-matrix VGPR layout: lanes 0-15 -> K=4k,4k+1 ; lanes 16-31 -> K=4k+2,4k+3).
__device__ __forceinline__ void bilerp_acc(const float* __restrict__ plane,
                                           size_t sOff, float u, float v,
                                           int chOff, float2 acc[8]) {
  float xg = (u + 1.0f) * 0.5f * (float)(RDIM - 1);
  float yg = (v + 1.0f) * 0.5f * (float)(RDIM - 1);
  float xf = fminf(fmaxf(floorf(xg), 0.0f), (float)(RDIM - 2));
  float yf = fminf(fmaxf(floorf(yg), 0.0f), (float)(RDIM - 2));
  int x0 = (int)xf, y0 = (int)yf;
  float wx = xg - xf, wy = yg - yf;
  float w00 = (1.0f - wx) * (1.0f - wy), w01 = wx * (1.0f - wy);
  float w10 = (1.0f - wx) * wy,          w11 = wx * wy;
  const float* p00 = plane + sOff + (size_t)(y0 * RDIM + x0) * CDIM;
  const float* p10 = p00 + (size_t)RDIM * CDIM;
#pragma unroll
  for (int k = 0; k < 8; ++k) {
    int o = 4 * k + chOff;
    float2 f00 = *(const float2*)(p00 + o);
    float2 f01 = *(const float2*)(p00 + CDIM + o);
    float2 f10 = *(const float2*)(p10 + o);
    float2 f11 = *(const float2*)(p10 + CDIM + o);
    acc[k].x += f00.x * w00 + f01.x * w01 + f10.x * w10 + f11.x * w11;
    acc[k].y += f00.y * w00 + f01.y * w01 + f10.y * w10 + f11.y * w11;
  }
}

// One 16(points) x 32(out) dense layer + bias + ReLU.
// Weights come pre-swizzled in B-operand layout: P[((k*2+h)*2+t)*16+n] =
// (W[4k+2h][16t+n], W[4k+2h+1][16t+n]) so each lane's B operand is ONE
// aligned ds_load_b64 into an even VGPR pair.  Two independent C chains
// (N-tile 0/1) are interleaved so consecutive WMMAs don't serialize.
__device__ __forceinline__ void mlp_layer(const float2* __restrict__ wp,
                                          const float* __restrict__ blds,
                                          const v2f a[8], float* __restrict__ dst,
                                          int n, int half) {
  const float2* wl = wp + half * 32 + n;
  v2f bb0[8], bb1[8];
#pragma unroll
  for (int k = 0; k < 8; ++k) {
    bb0[k] = *(const v2f*)(wl + k * 64);
    bb1[k] = *(const v2f*)(wl + k * 64 + 16);
  }
  float bias0 = blds[n], bias1 = blds[16 + n];
  v8f c0, c1;
#pragma unroll
  for (int i = 0; i < 8; ++i) { c0[i] = bias0; c1[i] = bias1; }
#pragma unroll
  for (int k = 0; k < 8; ++k) {
    c0 = __builtin_amdgcn_wmma_f32_16x16x4_f32(false, a[k], false, bb0[k],
                                               (short)0, c0, false, false);
    c1 = __builtin_amdgcn_wmma_f32_16x16x4_f32(false, a[k], false, bb1[k],
                                               (short)0, c1, false, false);
  }
#pragma unroll
  for (int i = 0; i < 8; ++i) {           // C layout: VGPR i -> row i + 8*half
    dst[(i + 8 * half) * 32 + n]      = fmaxf(c0[i], 0.0f);
    dst[(i + 8 * half) * 32 + 16 + n] = fmaxf(c1[i], 0.0f);
  }
}

// Re-read a [16pt][32ch] LDS tile into A-matrix layout (ds_load_b64 per k-step).
__device__ __forceinline__ void lds_to_A(const float* __restrict__ src,
                                         v2f a[8], int n, int half) {
#pragma unroll
  for (int k = 0; k < 8; ++k) {
    float2 v = *(const float2*)(src + n * 32 + 4 * k + 2 * half);
    a[k].x = v.x;
    a[k].y = v.y;
  }
}

__global__ __launch_bounds__(256) void decoders_kernel(
    const float* __restrict__ p, const float* __restrict__ bnd,
    const float* __restrict__ pxy, const float* __restrict__ pxz, const float* __restrict__ pyz,
    const float* __restrict__ cxy, const float* __restrict__ cxz, const float* __restrict__ cyz,
    const float* __restrict__ w0, const float* __restrict__ b0,
    const float* __restrict__ w1, const float* __restrict__ b1,
    const float* __restrict__ w_out, const float* __restrict__ b_out,
    const float* __restrict__ cw0, const float* __restrict__ cb0,
    const float* __restrict__ cw1, const float* __restrict__ cb1,
    const float* __restrict__ cw_out, const float* __restrict__ cb_out,
    float* __restrict__ out, int N) {
  __shared__ float2 sW0p[512], sW1p[512], sCW0p[512], sCW1p[512];
  __shared__ float sB0[32], sB1[32], sCB0[32], sCB1[32];
  __shared__ float sWout[32], sCWout[96], sBias2[4], sBnd[48];
  __shared__ float sBufS[8][512];
  __shared__ float sBufC[8][512];

  const int tid = threadIdx.x;
  // Pre-swizzle all four 32x32 weight matrices into B-operand layout.
  for (int i = tid; i < 512; i += 256) {
    int nn = i & 15, tt = (i >> 4) & 1, hh = (i >> 5) & 1, kk = (i >> 6) & 7;
    int row = 4 * kk + 2 * hh, col = 16 * tt + nn;
    sW0p[i]  = make_float2(w0[row * 32 + col],  w0[(row + 1) * 32 + col]);
    sW1p[i]  = make_float2(w1[row * 32 + col],  w1[(row + 1) * 32 + col]);
    sCW0p[i] = make_float2(cw0[row * 32 + col], cw0[(row + 1) * 32 + col]);
    sCW1p[i] = make_float2(cw1[row * 32 + col], cw1[(row + 1) * 32 + col]);
  }
  if (tid < 32) { sB0[tid] = b0[tid]; sB1[tid] = b1[tid]; sCB0[tid] = cb0[tid];
                  sCB1[tid] = cb1[tid]; sWout[tid] = w_out[tid]; }
  if (tid < 96) sCWout[tid] = cw_out[tid];
  if (tid < 48) sBnd[tid]  = bnd[tid];
  if (tid == 0) sBias2[0]  = b_out[0];
  if (tid < 3)  sBias2[1 + tid] = cb_out[tid];
  __syncthreads();

  const int lane = tid & 31;
  const int half = lane >> 4;
  const int n    = lane & 15;
  const int wib  = tid >> 5;
  float* bufS = &sBufS[wib][0];
  float* bufC = &sBufC[wib][0];

  const int groups     = (N + 15) >> 4;
  const int waveGlobal = blockIdx.x * 8 + wib;
  const int nWaves     = gridDim.x * 8;

  for (int g = waveGlobal; g < groups; g += nWaves) {
    int pt  = g * 16 + n;
    int ptc = pt < N ? pt : N - 1;          // clamp (keeps EXEC full for WMMA)

    float px = p[ptc * 3 + 0];
    float py = p[ptc * 3 + 1];
    float pz = p[ptc * 3 + 2];

    // segment select (argmax of inside) + valid mask
    int seg = -1;
#pragma unroll
    for (int s2 = 0; s2 < NSEG; ++s2) {
      const float* lo = &sBnd[s2 * 6];       // [lo0,lo1,lo2,hi0,hi1,hi2]
      bool in = (px > lo[0]) && (px < lo[3]) && (py > lo[1]) && (py < lo[4]) &&
                (pz > lo[2]) && (pz < lo[5]);
      if (in && seg < 0) seg = s2;
    }
    float m  = (seg >= 0) ? 1.0f : 0.0f;
    int   si = (seg >= 0) ? seg : 0;
    const float* lo = &sBnd[si * 6];
    float nx = (px - lo[0]) / (lo[3] - lo[0]) * 2.0f - 1.0f;
    float ny = (py - lo[1]) / (lo[4] - lo[1]) * 2.0f - 1.0f;
    float nz = (pz - lo[2]) / (lo[5] - lo[2]) * 2.0f - 1.0f;

    size_t sOff  = (size_t)si * (size_t)(RDIM * RDIM * CDIM);
    int    chOff = 2 * half;

    float2 fa[8], ca[8];
#pragma unroll
    for (int k = 0; k < 8; ++k) { fa[k] = make_float2(0.f, 0.f); ca[k] = make_float2(0.f, 0.f); }
    bilerp_acc(pxy, sOff, nx, ny, chOff, fa);
    bilerp_acc(pxz, sOff, nx, nz, chOff, fa);
    bilerp_acc(pyz, sOff, ny, nz, chOff, fa);
    bilerp_acc(cxy, sOff, nx, ny, chOff, ca);
    bilerp_acc(cxz, sOff, nx, nz, chOff, ca);
    bilerp_acc(cyz, sOff, ny, nz, chOff, ca);

    v2f a[8], ac[8];
#pragma unroll
    for (int k = 0; k < 8; ++k) {
      a[k].x  = fa[k].x * m; a[k].y  = fa[k].y * m;
      ac[k].x = ca[k].x * m; ac[k].y = ca[k].y * m;
    }

    // SDF trunk: 32 -> 32 -> 32 (ReLU), all via f32 WMMA
    mlp_layer(sW0p, sB0, a, bufS, n, half);
    lds_to_A(bufS, a, n, half);
    mlp_layer(sW1p, sB1, a, bufS, n, half);

    // Color trunk
    mlp_layer(sCW0p, sCB0, ac, bufC, n, half);
    lds_to_A(bufC, ac, n, half);
    mlp_layer(sCW1p, sCB1, ac, bufC, n, half);

    // Heads: half 0 -> (r, g) ; half 1 -> (b, sdf). Coalesced float2 stores.
    const float* hrow = bufS + n * 32;
    const float* crow = bufC + n * 32;
    float acc0 = 0.0f, acc1 = 0.0f;
    if (half == 0) {
#pragma unroll
      for (int i = 0; i < 32; ++i) {
        float hv = crow[i];
        acc0 += hv * sCWout[i * 3 + 0];
        acc1 += hv * sCWout[i * 3 + 1];
      }
      acc0 = sigmoidf_(acc0 + sBias2[1]);
      acc1 = sigmoidf_(acc1 + sBias2[2]);
    } else {
#pragma unroll
      for (int i = 0; i < 32; ++i) {
        acc0 += crow[i] * sCWout[i * 3 + 2];
        acc1 += hrow[i] * sWout[i];
      }
      acc0 = sigmoidf_(acc0 + sBias2[3]);
      acc1 = tanhf(acc1 + sBias2[0]);
    }
    if (pt < N) {
      *(float2*)(out + (size_t)pt * 4 + 2 * half) = make_float2(acc0, acc1);
    }
  }
}

extern "C" void kernel_launch(void* const* d_in, const int* in_sizes, int n_in,
                              void* d_out, int out_size, void* d_ws, size_t ws_size,
                              hipStream_t stream) {
  const float* p      = (const float*)d_in[0];
  const float* bnd    = (const float*)d_in[1];
  const float* pxy    = (const float*)d_in[2];
  const float* pxz    = (const float*)d_in[3];
  const float* pyz    = (const float*)d_in[4];
  const float* cxy    = (const float*)d_in[5];
  const float* cxz    = (const float*)d_in[6];
  const float* cyz    = (const float*)d_in[7];
  const float* w0     = (const float*)d_in[8];
  const float* b0     = (const float*)d_in[9];
  const float* w1     = (const float*)d_in[10];
  const float* b1     = (const float*)d_in[11];
  const float* w_out  = (const float*)d_in[12];
  const float* b_out  = (const float*)d_in[13];
  const float* cw0    = (const float*)d_in[14];
  const float* cb0    = (const float*)d_in[15];
  const float* cw1    = (const float*)d_in[16];
  const float* cb1    = (const float*)d_in[17];
  const float* cw_out = (const float*)d_in[18];
  const float* cb_out = (const float*)d_in[19];

  int N = in_sizes[0] / 3;
  int groups = (N + 15) / 16;
  int maxBlocks = (groups + 7) / 8;
  int nBlocks = maxBlocks < 2048 ? maxBlocks : 2048;

  decoders_kernel<<<nBlocks, 256, 0, stream>>>(
      p, bnd, pxy, pxz, pyz, cxy, cxz, cyz,
      w0, b0, w1, b1, w_out, b_out,
      cw0, cb0, cw1, cb1, cw_out, cb_out,
      (float*)d_out, N);
}